// GNN_30107720744960
// MI455X (gfx1250) — compile-verified
//
#include <hip/hip_runtime.h>

#define N_NODES 100000
#define N_EDGES 640000
#define DIM     128
#define N_LAYERS 4
#define BN_EPS  1e-5f

typedef unsigned int u32;
typedef __attribute__((ext_vector_type(2))) float v2f;
typedef __attribute__((ext_vector_type(8))) float v8f;
typedef __attribute__((ext_vector_type(4))) u32   v4u;
typedef __attribute__((ext_vector_type(8))) int   v8i;
typedef __attribute__((ext_vector_type(4))) int   v4i;

// ---------------------------------------------------------------------------
// Zero the 256-float stats buffer (sum[128] | sumsq[128])
// ---------------------------------------------------------------------------
__global__ void zero_stats_kernel(float* __restrict__ stats) {
    int t = threadIdx.x;
    if (t < 2 * DIM) stats[t] = 0.0f;
}

// ---------------------------------------------------------------------------
// z = (1 + eps[l]) * x     (float4-wide, fully streaming)
// ---------------------------------------------------------------------------
__global__ void scale_init_kernel(const float* __restrict__ x,
                                  float* __restrict__ z,
                                  const float* __restrict__ eps, int layer, int n4) {
    int i = blockIdx.x * blockDim.x + threadIdx.x;
    if (i >= n4) return;
    float sc = 1.0f + eps[layer];
    float4 v = ((const float4*)x)[i];
    v.x *= sc; v.y *= sc; v.z *= sc; v.w *= sc;
    ((float4*)z)[i] = v;
}

// ---------------------------------------------------------------------------
// Edge phase: msg = relu(x[src] + edge_attr); z[dst] += msg
// 32 lanes per edge, float4 per lane, native global f32 atomics for scatter.
// ---------------------------------------------------------------------------
__global__ void edge_kernel(const float* __restrict__ x,
                            const float* __restrict__ ea,
                            const long long* __restrict__ src,
                            const long long* __restrict__ dst,
                            float* __restrict__ z) {
    unsigned gid = blockIdx.x * blockDim.x + threadIdx.x;
    if (gid >= (unsigned)N_EDGES * 32u) return;
    unsigned e = gid >> 5;
    unsigned c = (gid & 31u) << 2;          // element offset 0..124
    long long s = src[e];
    long long d = dst[e];
    float4 xv = *(const float4*)(x + (size_t)s * DIM + c);
    float4 ev = *(const float4*)(ea + (size_t)e * DIM + c);
    float4 m;
    m.x = fmaxf(xv.x + ev.x, 0.0f);
    m.y = fmaxf(xv.y + ev.y, 0.0f);
    m.z = fmaxf(xv.z + ev.z, 0.0f);
    m.w = fmaxf(xv.w + ev.w, 0.0f);
    float* zp = z + (size_t)d * DIM + c;
    unsafeAtomicAdd(zp + 0, m.x);
    unsafeAtomicAdd(zp + 1, m.y);
    unsafeAtomicAdd(zp + 2, m.z);
    unsafeAtomicAdd(zp + 3, m.w);
}

// ---------------------------------------------------------------------------
// H = Z @ W (+ bias), f32 WMMA 16x16x4. One wave per 16-row strip, 8 column
// tiles of 16 -> 256 v_wmma per wave. The 128x128 f32 weight tile (64 KB) is
// staged into LDS by the Tensor Data Mover (one DMA descriptor issued by wave
// 0, completion via s_wait_tensorcnt, then block barrier). A seed store +
// memory clobber keep the compiler from undef-folding the LDS reads (the
// TDM's LDS writes are invisible to alias analysis). A fragments are
// software-pipelined v2f global loads. Optionally accumulates per-column
// sum / sum-of-squares for BatchNorm batch stats.
// ---------------------------------------------------------------------------
__launch_bounds__(256)
__global__ void gemm_wmma_kernel(const float* __restrict__ Z,
                                 const float* __restrict__ W,
                                 float* __restrict__ H,
                                 float* __restrict__ stats,
                                 const float* __restrict__ bias,
                                 int nStrips, int withStats) {
    __shared__ float wlds[DIM * DIM];       // 64 KB; ONLY LDS object -> offset 0

    int tid = threadIdx.x;
    if (tid < 32) {                         // wave 0 issues the TDM descriptor
        if (tid == 0) {
            // Seed store: makes wlds visibly stored-to so the B-operand loads
            // below cannot be folded to undef. It lands (s_wait_dscnt 0)
            // before the DMA overwrites it with real data.
            *(volatile float*)wlds = 0.0f;
        }
        __asm__ volatile("s_wait_dscnt 0" ::: "memory");

        unsigned long long ga = (unsigned long long)W;
        // D# group 0: count=1 | lds_addr | global_addr | type=2 ("image")
        v4u g0;
        g0.x = 1u;                                        // count=1, user mode
        g0.y = 0u;                                        // lds_addr = 0 (wlds)
        g0.z = (u32)(ga & 0xFFFFFFFFu);                   // global_addr[31:0]
        g0.w = (u32)((ga >> 32) & 0x01FFFFFFu)            // global_addr[56:32]
             | (2u << 30);                                // type = 2
        // D# group 1: data_size=4B, 128x128 tensor, 128x128 tile, stride 128
        v8i g1;
        g1[0] = (int)(2u << 16);            // workgroup_mask=0, data_size=2 (4B)
        g1[1] = (int)(128u << 16);          // tensor_dim0[15:0] = 128 (bits 63:48)
        g1[2] = (int)(128u << 16);          // tensor_dim0 hi=0 | tensor_dim1 lo=128
        g1[3] = (int)(128u << 16);          // tensor_dim1 hi=0 | tile_dim0=128
        g1[4] = 128;                        // tile_dim1=128, tile_dim2=0
        g1[5] = 128;                        // tensor_dim0_stride = 128 (low 32)
        g1[6] = 0;                          // stride hi | tensor_dim1_stride lo
        g1[7] = 0;                          // tensor_dim1_stride hi
        v4i gz4 = {0, 0, 0, 0};             // groups 2/3 unused (2-D tensor)
        v8i gz8 = {0, 0, 0, 0, 0, 0, 0, 0}; // extra group (clang-23 form), unused
        __builtin_amdgcn_tensor_load_to_lds(g0, g1, gz4, gz4, gz8, 0);
        __builtin_amdgcn_s_wait_tensorcnt(0);
    }
    __syncthreads();                        // W tile now resident in LDS
    __asm__ volatile("" ::: "memory");      // LDS was written behind clang's back

    int wave = tid >> 5;
    int lane = tid & 31;
    int strip = blockIdx.x * 8 + wave;      // 16 output rows per wave
    if (strip >= nStrips) return;           // wave-uniform: EXEC all-ones inside

    int half = lane >> 4;                   // 0: K=0,1  1: K=2,3 (A layout)
    int m    = lane & 15;                   // row (A) / column-in-tile (B,C,D)
    int r0   = strip * 16;
    // per-lane pointer to this lane's K-pairs in its A row
    const float* ap = Z + (size_t)(r0 + m) * DIM + 2 * half;

    v8f zero = {};
    v8f acc[8];
#pragma unroll
    for (int nt = 0; nt < 8; ++nt) acc[nt] = zero;

    // depth-2 A-fragment pipeline: a_cur for k, a_nxt for k+4.
    v2f a_cur = *(const v2f*)(ap + 0);
    v2f a_nxt = *(const v2f*)(ap + 4);

#pragma unroll 4
    for (int k = 0; k < DIM; k += 4) {
        v2f a = a_cur;
        a_cur = a_nxt;
        // Unconditional prefetch: for k >= 120 this overreads past the row,
        // but stays inside the d_ws allocation and the value is never used.
        a_nxt = *(const v2f*)(ap + k + 8);

        const float* wr0 = &wlds[(k + 2 * half + 0) * DIM + m];
        const float* wr1 = &wlds[(k + 2 * half + 1) * DIM + m];
#pragma unroll
        for (int nt = 0; nt < 8; ++nt) {
            v2f b;
            b.x = wr0[nt * 16];
            b.y = wr1[nt * 16];
            acc[nt] = __builtin_amdgcn_wmma_f32_16x16x4_f32(
                false, a, false, b, (short)0, acc[nt], false, false);
        }
    }

    // Epilogue: store H tile (+bias) and fold BN partial sums.
    // C/D layout: lane holds col = nt*16 + m, rows r0 + v + 8*half, v=0..7
#pragma unroll
    for (int nt = 0; nt < 8; ++nt) {
        int col = nt * 16 + m;
        float bb = (bias != nullptr) ? bias[col] : 0.0f;
        float* hp = H + (size_t)(r0 + 8 * half) * DIM + col;
        float s = 0.0f, sq = 0.0f;
#pragma unroll
        for (int v = 0; v < 8; ++v) {
            float val = acc[nt][v] + bb;
            hp[(size_t)v * DIM] = val;
            s  += val;
            sq += val * val;
        }
        if (withStats) {
            // lanes L and L^16 hold the same column (different rows): pair-reduce
            s  += __shfl_xor(s, 16);
            sq += __shfl_xor(sq, 16);
            if (half == 0) {
                unsafeAtomicAdd(&stats[col], s);
                unsafeAtomicAdd(&stats[DIM + col], sq);
            }
        }
    }
}

// ---------------------------------------------------------------------------
// x = relu((h - mu) * rsqrt(var + eps) * gamma + beta) + x   (in place)
// ---------------------------------------------------------------------------
__global__ void bn_relu_res_kernel(const float* __restrict__ H,
                                   float* __restrict__ x,
                                   const float* __restrict__ stats,
                                   const float* __restrict__ gamma,
                                   const float* __restrict__ beta,
                                   int layer, int n) {
    int i = blockIdx.x * blockDim.x + threadIdx.x;
    if (i >= n) return;
    int d = i & (DIM - 1);
    const float invN = 1.0f / (float)N_NODES;
    float mu  = stats[d] * invN;
    float var = stats[DIM + d] * invN - mu * mu;
    float rs  = rsqrtf(var + BN_EPS);
    float g   = gamma[layer * DIM + d];
    float b   = beta[layer * DIM + d];
    float h   = H[i];
    float v   = fmaxf((h - mu) * rs * g + b, 0.0f);
    x[i] = v + x[i];
}

// ---------------------------------------------------------------------------
// Host orchestration
// ---------------------------------------------------------------------------
extern "C" void kernel_launch(void* const* d_in, const int* in_sizes, int n_in,
                              void* d_out, int out_size, void* d_ws, size_t ws_size,
                              hipStream_t stream) {
    const float*     x_in  = (const float*)d_in[0];
    const long long* ei    = (const long long*)d_in[1];   // int64 [2, E]
    const float*     ea    = (const float*)d_in[2];
    const float*     Ws    = (const float*)d_in[3];       // [L, D, D]
    const float*     eps   = (const float*)d_in[4];       // [L]
    const float*     gamma = (const float*)d_in[5];       // [L, D]
    const float*     beta  = (const float*)d_in[6];       // [L, D]
    const float*     W_out = (const float*)d_in[7];       // [D, D]
    const float*     b_out = (const float*)d_in[8];       // [D]
    float*           out   = (float*)d_out;

    const size_t ND = (size_t)N_NODES * DIM;              // 12.8M floats
    float* ws    = (float*)d_ws;
    float* x_ws  = ws;                                    // current features
    float* z     = ws + ND;                               // (1+eps)*x + agg
    float* h     = ws + 2 * ND;                           // post-GEMM
    float* stats = ws + 3 * ND;                           // sum[128]|sumsq[128]

    (void)hipMemcpyAsync(x_ws, x_in, ND * sizeof(float),
                         hipMemcpyDeviceToDevice, stream);

    const long long* src = ei;
    const long long* dst = ei + N_EDGES;

    const int n4         = (int)(ND / 4);                 // 3.2M float4
    const int nStrips    = N_NODES / 16;                  // 6250 (exact)
    const int gemmBlocks = (nStrips + 7) / 8;             // 8 waves/block
    const int edgeBlocks = (N_EDGES * 32) / 256;          // 80000 (exact)

    for (int l = 0; l < N_LAYERS; ++l) {
        zero_stats_kernel<<<1, 256, 0, stream>>>(stats);
        scale_init_kernel<<<(n4 + 255) / 256, 256, 0, stream>>>(x_ws, z, eps, l, n4);
        edge_kernel<<<edgeBlocks, 256, 0, stream>>>(x_ws, ea, src, dst, z);
        gemm_wmma_kernel<<<gemmBlocks, 256, 0, stream>>>(
            z, Ws + (size_t)l * DIM * DIM, h, stats, nullptr, nStrips, 1);
        bn_relu_res_kernel<<<((int)ND + 255) / 256, 256, 0, stream>>>(
            h, x_ws, stats, gamma, beta, l, (int)ND);
    }

    // out = x @ W_out + b_out
    gemm_wmma_kernel<<<gemmBlocks, 256, 0, stream>>>(
        x_ws, W_out, out, stats, b_out, nStrips, 0);
}